// TeamPhaseGNN_55130200211969
// MI455X (gfx1250) — compile-verified
//
#include <hip/hip_runtime.h>

#define HID 64
#define NUM_GRAPHS 50
#define NEG_SLOPE 0.2f

typedef float v2f __attribute__((ext_vector_type(2)));
typedef float v8f __attribute__((ext_vector_type(8)));

// ---- order-preserving float <-> uint keys (for atomic segment max) ----
__device__ __forceinline__ unsigned ordKey(float f) {
    unsigned u = __float_as_uint(f);
    return (u >> 31) ? ~u : (u | 0x80000000u);
}
__device__ __forceinline__ float keyOrd(unsigned k) {
    return __uint_as_float((k >> 31) ? (k ^ 0x80000000u) : ~k);
}
__device__ __forceinline__ float lrelu(float v) {
    return v > 0.f ? v : NEG_SLOPE * v;
}

__global__ void fill_u32_kernel(unsigned* __restrict__ p, unsigned v, long n) {
    long i = (long)blockIdx.x * blockDim.x + threadIdx.x;
    if (i < n) p[i] = v;
}

// ---- Y[M,N] = X[M,K] @ W[K,N] via V_WMMA_F32_16X16X4_F32, one wave / 16x16 tile ----
__global__ void gemm_f32_wmma(const float* __restrict__ X, const float* __restrict__ W,
                              float* __restrict__ Y, int M, int K, int N) {
    int wave = blockIdx.x * (blockDim.x >> 5) + (threadIdx.x >> 5);
    int lane = threadIdx.x & 31;
    int ntiles = N >> 4;
    int mtiles = M >> 4;
    if (wave >= ntiles * mtiles) return;             // wave-uniform: EXEC all-1s inside
    int m0 = (wave / ntiles) << 4;
    int n0 = (wave % ntiles) << 4;
    int half = lane >> 4;                            // which K-half of the 4-wide slab
    int l    = lane & 15;                            // M index (A) / N index (B,C,D)

    // A fragment: lane holds (M=m0+l, K=k0+2*half .. +1) -> contiguous b64 load
    const float* ap = X + (long)(m0 + l) * K + half * 2;
    // B fragment: lane holds (K=k0+2*half.., N=n0+l); rows stride N
    const float* bp = W + (long)(half * 2) * N + n0 + l;
    const long   bstep = 4L * N;

    v8f c = {0.f, 0.f, 0.f, 0.f, 0.f, 0.f, 0.f, 0.f};
#pragma unroll 4
    for (int k0 = 0; k0 < K; k0 += 4) {
        v2f a = *(const v2f*)ap;
        v2f b;
        b.x = bp[0];
        b.y = bp[N];
        c = __builtin_amdgcn_wmma_f32_16x16x4_f32(false, a, false, b, (short)0, c,
                                                  false, false);
        ap += 4;
        bp += bstep;
    }
    float* yrow = Y + (long)(m0 + half * 8) * N + n0 + l;   // C/D layout
#pragma unroll
    for (int r = 0; r < 8; ++r) yrow[(long)r * N] = c[r];
}

// ---- attention logits: one wave per edge; lanes cover H*HID channels ----
template <int H>
__global__ void edge_logits_kernel(const int* __restrict__ ei, int E, int ET,
                                   const float* __restrict__ xl,
                                   const float* __restrict__ xr,
                                   const float* __restrict__ att,
                                   float* __restrict__ logits,
                                   unsigned* __restrict__ mkey) {
    int e = blockIdx.x * (blockDim.x >> 5) + (threadIdx.x >> 5);
    int lane = threadIdx.x & 31;
    if (e >= ET) return;
    e = __builtin_amdgcn_readfirstlane(e);           // uniform -> scalar loads of s,d
    int s, d;
    if (e < E) { s = ei[e]; d = ei[E + e]; } else { s = d = e - E; }
    const int HC = H * HID;                          // 128 (H=2) or 64 (H=1)
    int head = (H == 2) ? (lane >> 4) : 0;
    float p = 0.f;
    if (H == 2) {                                    // float4 per lane, coalesced b128
        float4 a = ((const float4*)(xl + (long)s * HC))[lane];
        float4 b = ((const float4*)(xr + (long)d * HC))[lane];
        float4 t = ((const float4*)att)[lane];
        p = t.x * lrelu(a.x + b.x) + t.y * lrelu(a.y + b.y) +
            t.z * lrelu(a.z + b.z) + t.w * lrelu(a.w + b.w);
    } else {                                         // float2 per lane, coalesced b64
        float2 a = ((const float2*)(xl + (long)s * HC))[lane];
        float2 b = ((const float2*)(xr + (long)d * HC))[lane];
        float2 t = ((const float2*)att)[lane];
        p = t.x * lrelu(a.x + b.x) + t.y * lrelu(a.y + b.y);
    }
    const int RED = (H == 2) ? 16 : 32;
#pragma unroll
    for (int m = 1; m < RED; m <<= 1) p += __shfl_xor(p, m, 32);
    if ((lane & (RED - 1)) == 0) {
        logits[(long)e * H + head] = p;
        atomicMax(&mkey[(long)d * H + head], ordKey(p));
    }
}

// ---- a = exp(logit - m[dst]); denom[dst] += a ----
__global__ void edge_softnum_kernel(const int* __restrict__ ei, int E, int ET, int H,
                                    const unsigned* __restrict__ mkey,
                                    float* __restrict__ alpha,
                                    float* __restrict__ denom) {
    long i = (long)blockIdx.x * blockDim.x + threadIdx.x;
    if (i >= (long)ET * H) return;
    int e = (int)(i / H), h = (int)(i % H);
    int d = (e < E) ? ei[E + e] : (e - E);
    float m = keyOrd(mkey[(long)d * H + h]);
    float a = expf(alpha[i] - m);
    alpha[i] = a;
    atomicAdd(&denom[(long)d * H + h], a);
}

// ---- out[dst] += (alpha/denom) * xl[src], one wave per edge ----
template <int H>
__global__ void edge_aggregate_kernel(const int* __restrict__ ei, int E, int ET,
                                      const float* __restrict__ xl,
                                      const float* __restrict__ alpha,
                                      const float* __restrict__ denom,
                                      float* __restrict__ out) {
    int e = blockIdx.x * (blockDim.x >> 5) + (threadIdx.x >> 5);
    int lane = threadIdx.x & 31;
    if (e >= ET) return;
    e = __builtin_amdgcn_readfirstlane(e);
    int s, d;
    if (e < E) { s = ei[e]; d = ei[E + e]; } else { s = d = e - E; }
    const int HC = H * HID;
    int head = (H == 2) ? (lane >> 4) : 0;
    float w = alpha[(long)e * H + head] / (denom[(long)d * H + head] + 1e-16f);
    float* orow = out + (long)d * HC;
    if (H == 2) {
        float4 a = ((const float4*)(xl + (long)s * HC))[lane];
        int c = lane * 4;
        atomicAdd(&orow[c + 0], w * a.x);
        atomicAdd(&orow[c + 1], w * a.y);
        atomicAdd(&orow[c + 2], w * a.z);
        atomicAdd(&orow[c + 3], w * a.w);
    } else {
        float2 a = ((const float2*)(xl + (long)s * HC))[lane];
        int c = lane * 2;
        atomicAdd(&orow[c + 0], w * a.x);
        atomicAdd(&orow[c + 1], w * a.y);
    }
}

__global__ void bias_relu_kernel(float* __restrict__ h, const float* __restrict__ b,
                                 int hcMask, long n) {
    long i = (long)blockIdx.x * blockDim.x + threadIdx.x;
    if (i >= n) return;
    float v = h[i] + b[(int)i & hcMask];
    h[i] = v > 0.f ? v : 0.f;
}

// ---- graph pooling: mean-sum / max / count ----
__global__ void pool_accum_kernel(const float* __restrict__ h, const int* __restrict__ batch,
                                  float* __restrict__ sum, unsigned* __restrict__ maxk,
                                  float* __restrict__ cnt, int N, int C) {
    long i = (long)blockIdx.x * blockDim.x + threadIdx.x;
    if (i >= (long)N * C) return;
    int n = (int)(i / C), c = (int)(i % C);
    int g = batch[n];
    float v = h[i];
    atomicAdd(&sum[(long)g * C + c], v);
    atomicMax(&maxk[(long)g * C + c], ordKey(v));
    if (c == 0) atomicAdd(&cnt[g], 1.0f);
}

__global__ void pool_finalize_kernel(const float* __restrict__ sum,
                                     const unsigned* __restrict__ maxk,
                                     const float* __restrict__ cnt,
                                     float* __restrict__ hg, int G, int C) {
    int i = blockIdx.x * blockDim.x + threadIdx.x;
    if (i >= G * C) return;
    int g = i / C, c = i % C;
    float ct = cnt[g];
    ct = ct > 1.f ? ct : 1.f;
    hg[(long)g * 2 * C + c] = sum[i] / ct;
    float mx = keyOrd(maxk[i]);
    hg[(long)g * 2 * C + C + c] = mx > 0.f ? mx : 0.f;   // max(hg_max, 0)
}

__global__ void proj_relu_kernel(const float* __restrict__ hg, const float* __restrict__ w,
                                 const float* __restrict__ b, float* __restrict__ emb, int G) {
    int i = blockIdx.x * blockDim.x + threadIdx.x;
    if (i >= G * 64) return;
    int g = i / 64, j = i % 64;
    float acc = b[j];
    for (int k = 0; k < 128; ++k) acc += hg[g * 128 + k] * w[k * 64 + j];
    emb[i] = acc > 0.f ? acc : 0.f;
}

__global__ void fc_out_kernel(const float* __restrict__ emb, const float* __restrict__ w,
                              const float* __restrict__ b, float* __restrict__ logits, int G) {
    int i = blockIdx.x * blockDim.x + threadIdx.x;
    if (i >= G * 10) return;
    int g = i / 10, q = i % 10;
    float acc = b[q];
    for (int j = 0; j < 64; ++j) acc += emb[g * 64 + j] * w[j * 10 + q];
    logits[i] = acc;
}

static inline int cdiv(long a, long b) { return (int)((a + b - 1) / b); }

extern "C" void kernel_launch(void* const* d_in, const int* in_sizes, int n_in,
                              void* d_out, int out_size, void* d_ws, size_t ws_size,
                              hipStream_t stream) {
    const float* x      = (const float*)d_in[0];
    const int*   ei     = (const int*)d_in[1];
    const int*   batch  = (const int*)d_in[2];
    const float* wl1    = (const float*)d_in[3];
    const float* wr1    = (const float*)d_in[4];
    const float* att1   = (const float*)d_in[5];
    const float* b1     = (const float*)d_in[6];
    const float* wl2    = (const float*)d_in[7];
    const float* wr2    = (const float*)d_in[8];
    const float* att2   = (const float*)d_in[9];
    const float* b2     = (const float*)d_in[10];
    const float* w_proj = (const float*)d_in[11];
    const float* b_proj = (const float*)d_in[12];
    const float* w_fc   = (const float*)d_in[13];
    const float* b_fc   = (const float*)d_in[14];

    const int N  = in_sizes[0] / 64;     // 50000
    const int E  = in_sizes[1] / 2;      // 1600000
    const int ET = E + N;                // + self loops
    const int G  = NUM_GRAPHS;

    // ---- workspace layout (floats), with layer-2 reuse of layer-1 regions ----
    float* ws = (float*)d_ws;
    long off = 0;
    float* R0 = ws + off; off += (long)N * 128;   // xl1 ; later [xl2 | out2/h2]
    float* R1 = ws + off; off += (long)N * 128;   // xr1 ; later xr2
    float* R2 = ws + off; off += (long)N * 128;   // out1 / h1
    float* AL = ws + off; off += (long)ET * 2;    // logits -> alpha
    unsigned* MK = (unsigned*)(ws + off); off += (long)N * 2;   // seg-max keys
    float* DN = ws + off; off += (long)N * 2;     // softmax denom
    float* PS = ws + off; off += (long)G * 64;    // pool sum
    unsigned* PM = (unsigned*)(ws + off); off += (long)G * 64;  // pool max keys
    float* PC = ws + off; off += G;               // pool counts
    float* HG = ws + off; off += (long)G * 128;   // concat(mean,max)

    float* xl2 = R0;
    float* h2  = R0 + (long)N * 64;
    float* xr2 = R1;

    float* out_logits = (float*)d_out;                // [50,10]
    float* out_emb    = (float*)d_out + (long)G * 10; // [50,64]

    const int BLK = 256;                 // 8 wave32s
    const int WPB = BLK / 32;

    // ===== Layer 1: xl1 = x@wl1, xr1 = x@wr1  (WMMA f32) =====
    {
        int tiles = (N / 16) * (128 / 16);
        gemm_f32_wmma<<<cdiv(tiles, WPB), BLK, 0, stream>>>(x, wl1, R0, N, 64, 128);
        gemm_f32_wmma<<<cdiv(tiles, WPB), BLK, 0, stream>>>(x, wr1, R1, N, 64, 128);
    }
    // zero out1 accumulator, seg-max keys, denoms (MK/DN are contiguous)
    fill_u32_kernel<<<cdiv((long)N * 128, BLK), BLK, 0, stream>>>((unsigned*)R2, 0u, (long)N * 128);
    fill_u32_kernel<<<cdiv((long)N * 4, BLK), BLK, 0, stream>>>((unsigned*)MK, 0u, (long)N * 4);

    edge_logits_kernel<2><<<cdiv(ET, WPB), BLK, 0, stream>>>(ei, E, ET, R0, R1, att1, AL, MK);
    edge_softnum_kernel<<<cdiv((long)ET * 2, BLK), BLK, 0, stream>>>(ei, E, ET, 2, MK, AL, DN);
    edge_aggregate_kernel<2><<<cdiv(ET, WPB), BLK, 0, stream>>>(ei, E, ET, R0, AL, DN, R2);
    bias_relu_kernel<<<cdiv((long)N * 128, BLK), BLK, 0, stream>>>(R2, b1, 127, (long)N * 128);

    // ===== Layer 2: xl2 = h1@wl2, xr2 = h1@wr2 =====
    {
        int tiles = (N / 16) * (64 / 16);
        gemm_f32_wmma<<<cdiv(tiles, WPB), BLK, 0, stream>>>(R2, wl2, xl2, N, 128, 64);
        gemm_f32_wmma<<<cdiv(tiles, WPB), BLK, 0, stream>>>(R2, wr2, xr2, N, 128, 64);
    }
    fill_u32_kernel<<<cdiv((long)N * 64, BLK), BLK, 0, stream>>>((unsigned*)h2, 0u, (long)N * 64);
    fill_u32_kernel<<<cdiv((long)N * 4, BLK), BLK, 0, stream>>>((unsigned*)MK, 0u, (long)N * 4);

    edge_logits_kernel<1><<<cdiv(ET, WPB), BLK, 0, stream>>>(ei, E, ET, xl2, xr2, att2, AL, MK);
    edge_softnum_kernel<<<cdiv((long)ET, BLK), BLK, 0, stream>>>(ei, E, ET, 1, MK, AL, DN);
    edge_aggregate_kernel<1><<<cdiv(ET, WPB), BLK, 0, stream>>>(ei, E, ET, xl2, AL, DN, h2);
    bias_relu_kernel<<<cdiv((long)N * 64, BLK), BLK, 0, stream>>>(h2, b2, 63, (long)N * 64);

    // ===== Pooling (mean + max per graph); PS/PM/PC contiguous =====
    fill_u32_kernel<<<cdiv((long)(G * 129), BLK), BLK, 0, stream>>>((unsigned*)PS, 0u, (long)G * 129);
    pool_accum_kernel<<<cdiv((long)N * 64, BLK), BLK, 0, stream>>>(h2, batch, PS, PM, PC, N, 64);
    pool_finalize_kernel<<<cdiv(G * 64, BLK), BLK, 0, stream>>>(PS, PM, PC, HG, G, 64);

    // ===== proj + fc =====
    proj_relu_kernel<<<cdiv(G * 64, BLK), BLK, 0, stream>>>(HG, w_proj, b_proj, out_emb, G);
    fc_out_kernel<<<cdiv(G * 10, 64), 64, 0, stream>>>(out_emb, w_fc, b_fc, out_logits, G);
}